// FCOSLoss_V2_18270790877917
// MI455X (gfx1250) — compile-verified
//
#include <hip/hip_runtime.h>
#include <cstdint>

#define N0    25600   // 160x160, stride 4
#define N1    6400    // 80x80,   stride 8
#define N2    1600    // 40x40,   stride 16
#define NTOT  33600
#define BATCH 16
#define NBOX  64
#define TPB   256
#define BLOCKS_PER_IMG ((NTOT + TPB - 1) / TPB)   // 132
#define NPARTIAL (BLOCKS_PER_IMG * BATCH)         // 2112

typedef __attribute__((ext_vector_type(2))) float v2f;
typedef __attribute__((ext_vector_type(8))) float v8f;

__device__ __forceinline__ float softplusf(float a) {
    // log(1 + e^a), numerically stable
    return fmaxf(a, 0.0f) + log1pf(__expf(-fabsf(a)));
}

// Exact wave32 sum-reduction of two f32 values via V_WMMA_F32_16X16X4_F32.
// A[m][k] packs (a,b) of lanes m and m+16; B is a 0/1 selector so that
// D[m][0] = a(m)+a(m+16), D[m][1] = b(m)+b(m+16). Per-lane sum of the 8
// D VGPRs + one xor-16 shuffle completes the reduction. f32*1.0 is exact,
// accumulation is f32 -> numerically a plain sum. EXEC must be all ones:
// callers invoke this unconditionally from every thread of the block.
__device__ __forceinline__ void wave_reduce_pair(float& a, float& b) {
#if __has_builtin(__builtin_amdgcn_wmma_f32_16x16x4_f32)
    const int lane = threadIdx.x & 31;
    v2f A; A.x = a; A.y = b;
    v2f B;
    B.x = ((lane & 15) == 0) ? 1.0f : 0.0f;   // B[0][n]/B[2][n] = (n==0)
    B.y = ((lane & 15) == 1) ? 1.0f : 0.0f;   // B[1][n]/B[3][n] = (n==1)
    v8f C = {};
    v8f D = __builtin_amdgcn_wmma_f32_16x16x4_f32(
        /*neg_a=*/false, A, /*neg_b=*/false, B,
        /*c_mod=*/(short)0, C, /*reuse_a=*/false, /*reuse_b=*/false);
    float s = ((D[0] + D[1]) + (D[2] + D[3])) + ((D[4] + D[5]) + (D[6] + D[7]));
    s += __shfl_xor(s, 16);    // lane0: full a-sum, lane1: full b-sum
    a = __shfl(s, 0);
    b = __shfl(s, 1);
#else
    #pragma unroll
    for (int off = 16; off > 0; off >>= 1) {
        a += __shfl_xor(a, off);
        b += __shfl_xor(b, off);
    }
#endif
}

__global__ __launch_bounds__(TPB) void fcos_main(
    const float* __restrict__ cls0, const float* __restrict__ cls1,
    const float* __restrict__ cls2, const float* __restrict__ reg0,
    const float* __restrict__ reg1, const float* __restrict__ reg2,
    const float* __restrict__ gt,   float* __restrict__ partial)
{
    __shared__ float4 sbox[NBOX];     // 64 boxes (x0,y0,x1,y1), 1 KB
    __shared__ float  sarea[NBOX];
    __shared__ float  swave[TPB / 32][3];

    const int tid = threadIdx.x;
    const int b   = blockIdx.y;
    const int n   = blockIdx.x * TPB + tid;

    // --- CDNA5 async copy: 64 boxes (1 KB) global -> LDS, ASYNCcnt-tracked ---
    if (tid < NBOX) {
        const float* g = gt + ((size_t)b * NBOX + (size_t)tid) * 4;   // 16B/lane
        unsigned lds_addr = (unsigned)(uintptr_t)(&sbox[tid]);        // flat[31:0] == LDS offset
        asm volatile("global_load_async_to_lds_b128 %0, %1, off"
                     :: "v"(lds_addr), "v"(g) : "memory");
    }
    asm volatile("s_wait_asynccnt 0" ::: "memory");
    __syncthreads();

    if (tid < NBOX) {
        float4 bb = sbox[tid];
        sarea[tid] = (bb.z - bb.x) * (bb.w - bb.y);
    }
    __syncthreads();

    float cls_acc = 0.0f, reg_acc = 0.0f, pos_acc = 0.0f;

    if (n < NTOT) {
        float x, y, low, high, z, pl, pt, pr, pb;
        if (n < N0) {
            int r = n / 160, c = n - r * 160;
            x = (c + 0.5f) * 4.0f;  y = (r + 0.5f) * 4.0f;
            low = -1.0f; high = 64.0f;
            z = cls0[(size_t)b * N0 + n];
            const float* rp = reg0 + (size_t)b * 4 * N0 + n;
            pl = rp[0]; pt = rp[N0]; pr = rp[2 * N0]; pb = rp[3 * N0];
        } else if (n < N0 + N1) {
            int m = n - N0;
            int r = m / 80, c = m - r * 80;
            x = (c + 0.5f) * 8.0f;  y = (r + 0.5f) * 8.0f;
            low = 64.0f; high = 128.0f;
            z = cls1[(size_t)b * N1 + m];
            const float* rp = reg1 + (size_t)b * 4 * N1 + m;
            pl = rp[0]; pt = rp[N1]; pr = rp[2 * N1]; pb = rp[3 * N1];
        } else {
            int m = n - (N0 + N1);
            int r = m / 40, c = m - r * 40;
            x = (c + 0.5f) * 16.0f; y = (r + 0.5f) * 16.0f;
            low = 128.0f; high = 99999.0f;
            z = cls2[(size_t)b * N2 + m];
            const float* rp = reg2 + (size_t)b * 4 * N2 + m;
            pl = rp[0]; pt = rp[N2]; pr = rp[2 * N2]; pb = rp[3 * N2];
        }

        const float INF = __builtin_inff();
        float minA = INF;
        int   idx  = 0;   // jnp.argmin of all-inf -> 0; strict '<' keeps first tie
        #pragma unroll 8
        for (int j = 0; j < NBOX; ++j) {
            float4 bb = sbox[j];                    // same-address LDS broadcast
            float l  = x - bb.x,  t = y - bb.y;
            float r2 = bb.z - x,  d = bb.w - y;
            float mn = fminf(fminf(l, t), fminf(r2, d));
            float mx = fmaxf(fmaxf(l, t), fmaxf(r2, d));
            bool ok  = (mn > 0.0f) && (mx > low) && (mx < high);
            float a  = ok ? sarea[j] : INF;
            bool take = a < minA;
            minA = take ? a : minA;
            idx  = take ? j : idx;
        }
        bool pos = minA < INF;

        float4 mb = sbox[idx];
        float tl = x - mb.x, tt = y - mb.y, tr = mb.z - x, tb2 = mb.w - y;

        // IoU / GIoU terms (shared)
        float w_i = fminf(pl, tl) + fminf(pr, tr);
        float h_i = fminf(pt, tt) + fminf(pb, tb2);
        float a_i = fmaxf(w_i, 0.0f) * fmaxf(h_i, 0.0f);
        float a_p = (pl + pr) * (pt + pb);
        float a_t = (tl + tr) * (tt + tb2);
        float a_u = a_p + a_t - a_i + 1e-7f;
        float iou = a_i / a_u;                       // unclipped (giou path)
        float w_e = fmaxf(pl, tl) + fmaxf(pr, tr);
        float h_e = fmaxf(pt, tt) + fmaxf(pb, tb2);
        float a_e = fmaxf(w_e, 0.0f) * fmaxf(h_e, 0.0f) + 1e-7f;
        float gl  = 1.0f - (iou - (a_e - a_u) / a_e);

        if (pos) { reg_acc = gl; pos_acc = 1.0f; }

        // varifocal (single logit channel), gt_iou = clipped IoU if positive
        float gtv  = pos ? fminf(fmaxf(iou, 0.0f), 1.0f) : 0.0f;
        float sp_p = softplusf(z);    // = -log_sigmoid(-z)
        float sp_n = softplusf(-z);   // = -log_sigmoid(z)
        float prob = 1.0f / (1.0f + __expf(-z));
        cls_acc = (gtv > 0.0f)
                    ? gtv * (gtv * sp_n + (1.0f - gtv) * sp_p)
                    : 0.75f * prob * prob * sp_p;
    }

    // --- wave32 reduction: WMMA for (cls,reg), butterfly for pos count ---
    wave_reduce_pair(cls_acc, reg_acc);
    #pragma unroll
    for (int off = 16; off > 0; off >>= 1) pos_acc += __shfl_xor(pos_acc, off);

    const int lane = tid & 31, wv = tid >> 5;
    if (lane == 0) {
        swave[wv][0] = cls_acc; swave[wv][1] = reg_acc; swave[wv][2] = pos_acc;
    }
    __syncthreads();
    if (tid == 0) {
        float c = 0.0f, r = 0.0f, p = 0.0f;
        #pragma unroll
        for (int w = 0; w < TPB / 32; ++w) {
            c += swave[w][0]; r += swave[w][1]; p += swave[w][2];
        }
        const int blk = blockIdx.y * gridDim.x + blockIdx.x;
        partial[blk * 3 + 0] = c;
        partial[blk * 3 + 1] = r;
        partial[blk * 3 + 2] = p;
    }
}

__global__ __launch_bounds__(TPB) void fcos_final(
    const float* __restrict__ partial, float* __restrict__ out)
{
    __shared__ float swave[TPB / 32][3];
    const int tid = threadIdx.x;
    float c = 0.0f, r = 0.0f, p = 0.0f;
    for (int i = tid; i < NPARTIAL; i += TPB) {      // fixed order: deterministic
        c += partial[i * 3 + 0];
        r += partial[i * 3 + 1];
        p += partial[i * 3 + 2];
    }
    wave_reduce_pair(c, r);
    #pragma unroll
    for (int off = 16; off > 0; off >>= 1) p += __shfl_xor(p, off);

    if ((tid & 31) == 0) {
        swave[tid >> 5][0] = c; swave[tid >> 5][1] = r; swave[tid >> 5][2] = p;
    }
    __syncthreads();
    if (tid == 0) {
        float C = 0.0f, R = 0.0f, P = 0.0f;
        #pragma unroll
        for (int w = 0; w < TPB / 32; ++w) {
            C += swave[w][0]; R += swave[w][1]; P += swave[w][2];
        }
        float navg = fmaxf(1.0f, P / (float)BATCH);
        out[0] = (C + R) / navg;
        out[1] = C / navg;
        out[2] = R / navg;
    }
}

extern "C" void kernel_launch(void* const* d_in, const int* in_sizes, int n_in,
                              void* d_out, int out_size, void* d_ws, size_t ws_size,
                              hipStream_t stream) {
    (void)in_sizes; (void)n_in; (void)out_size; (void)ws_size;
    const float* cls0 = (const float*)d_in[0];
    const float* cls1 = (const float*)d_in[1];
    const float* cls2 = (const float*)d_in[2];
    const float* reg0 = (const float*)d_in[3];
    const float* reg1 = (const float*)d_in[4];
    const float* reg2 = (const float*)d_in[5];
    const float* gt   = (const float*)d_in[6];
    float* partial = (float*)d_ws;          // 2112 * 3 floats = 25 KB
    float* out     = (float*)d_out;

    dim3 grid(BLOCKS_PER_IMG, BATCH);
    fcos_main<<<grid, TPB, 0, stream>>>(cls0, cls1, cls2, reg0, reg1, reg2, gt, partial);
    fcos_final<<<1, TPB, 0, stream>>>(partial, out);
}